// AttnDecoderRNN_49581102465492
// MI455X (gfx1250) — compile-verified
//
#include <hip/hip_runtime.h>
#include <hip/hip_bf16.h>

// ---------------------------------------------------------------------------
// AttnDecoderRNN step for MI455X (gfx1250, wave32).
// B=2048, L=256, H=512, O=1.
// GEMMs (attn_logits / comb / GRU gi,gh) use V_WMMA_F32_16X16X4_F32 (fp32),
// NT output tiles per wave sharing one A fragment (amortizes A loads and
// loop overhead). Concat-K handled as two clean loop segments -> no cndmask,
// incremental addressing, EXEC all-1s at every WMMA.
// The 1 GiB encoder_outputs stream (roofline anchor, ~46us @ 23.3 TB/s) is a
// coalesced float4 nontemporal VALU reduction (M=1 per batch -> no WMMA reuse).
// ---------------------------------------------------------------------------

#define Bsz 2048
#define Lsz 256
#define Hsz 512

typedef __attribute__((ext_vector_type(2))) float v2f;
typedef __attribute__((ext_vector_type(4))) float v4f;
typedef __attribute__((ext_vector_type(8))) float v8f;

// ---------------------------------------------------------------------------
// Kernel 1: embedded = x*emb_W + emb_b, then LayerNorm over H (biased var).
// ---------------------------------------------------------------------------
__global__ __launch_bounds__(128) void embed_ln_kernel(
    const float* __restrict__ x, const float* __restrict__ W,
    const float* __restrict__ bias, float* __restrict__ out /* B x H */) {
  __shared__ float sSum[4], sSq[4];
  const int b = blockIdx.x;
  const int t = threadIdx.x;
  const float xv = x[b];

  float4 w  = ((const float4*)W)[t];
  float4 bb = ((const float4*)bias)[t];
  float4 e;
  e.x = fmaf(xv, w.x, bb.x);
  e.y = fmaf(xv, w.y, bb.y);
  e.z = fmaf(xv, w.z, bb.z);
  e.w = fmaf(xv, w.w, bb.w);

  float s = e.x + e.y + e.z + e.w;
  float q = e.x * e.x + e.y * e.y + e.z * e.z + e.w * e.w;
#pragma unroll
  for (int o = 16; o; o >>= 1) {
    s += __shfl_xor(s, o, 32);
    q += __shfl_xor(q, o, 32);
  }
  const int lane = t & 31, wv = t >> 5;
  if (lane == 0) { sSum[wv] = s; sSq[wv] = q; }
  __syncthreads();
  const float ts = sSum[0] + sSum[1] + sSum[2] + sSum[3];
  const float tq = sSq[0] + sSq[1] + sSq[2] + sSq[3];
  const float mu  = ts * (1.0f / (float)Hsz);
  const float var = tq * (1.0f / (float)Hsz) - mu * mu;
  const float inv = rsqrtf(var + 1e-5f);

  float4 o4;
  o4.x = (e.x - mu) * inv;
  o4.y = (e.y - mu) * inv;
  o4.z = (e.z - mu) * inv;
  o4.w = (e.w - mu) * inv;
  ((float4*)(out + (size_t)b * Hsz))[t] = o4;
}

// ---------------------------------------------------------------------------
// Kernel 2: fp32 WMMA GEMM  C[M,N] = [A0 | A1][M, K0+K1] @ Bm[K,N] + bias(N)
// (optional ReLU).  Row-major everywhere.
// Block = 128 threads = 4 waves; wave w computes a 16 x (16*NT) strip at
// rows [blockIdx.y*64 + w*16), cols [blockIdx.x*16*NT).
//
// Fragment layouts per CDNA5 ISA 7.12.2 (V_WMMA_F32_16X16X4_F32):
//   A 16x4 : lane<16 -> (M=lane, K=k+0/k+1 in v[0]/v[1]); lane>=16 -> K=k+2/k+3
//   B 4x16 : lane&15 = N col; v[0]=row k+2*(lane>>4), v[1]=next row
//   C 16x16: vgpr r, row = r + 8*(lane>>4), col = lane&15
// ---------------------------------------------------------------------------
template <int NT>
__global__ __launch_bounds__(128) void gemm_wmma_f32_kernel(
    const float* __restrict__ A0, const float* __restrict__ A1, int K0, int K1,
    const float* __restrict__ Bm, const float* __restrict__ bias,
    float* __restrict__ C, int N, int relu) {
  const int lane = threadIdx.x & 31;
  const int wave = threadIdx.x >> 5;
  const int rowBase = blockIdx.y * 64 + wave * 16;
  const int colBase = blockIdx.x * (16 * NT);

  const int halfSel = lane >> 4;   // 0: lanes 0-15, 1: lanes 16-31
  const int mlane   = lane & 15;
  const int n = colBase + mlane;

  v8f acc[NT];
#pragma unroll
  for (int t = 0; t < NT; ++t) acc[t] = (v8f){};

  // B pointer walks rows 0..K-1 across BOTH segments (concat in K).
  const float* bptr = Bm + (size_t)(2 * halfSel) * N + n;

  // ---- segment 0: A0, K0 steps ----
  const float* aptr = A0 + (size_t)(rowBase + mlane) * K0 + 2 * halfSel;
  for (int k = 0; k < K0; k += 4) {
    const v2f a = *(const v2f*)aptr;           // 8B-aligned global_load_b64
#pragma unroll
    for (int t = 0; t < NT; ++t) {
      v2f bf;
      bf[0] = bptr[16 * t];
      bf[1] = bptr[N + 16 * t];
      acc[t] = __builtin_amdgcn_wmma_f32_16x16x4_f32(
          false, a, false, bf, (short)0, acc[t], false, false);
    }
    aptr += 4;
    bptr += (size_t)4 * N;
  }

  // ---- segment 1: A1, K1 steps (skipped when K1 == 0) ----
  aptr = A1 + (size_t)(rowBase + mlane) * K1 + 2 * halfSel;
  for (int k = 0; k < K1; k += 4) {
    const v2f a = *(const v2f*)aptr;
#pragma unroll
    for (int t = 0; t < NT; ++t) {
      v2f bf;
      bf[0] = bptr[16 * t];
      bf[1] = bptr[N + 16 * t];
      acc[t] = __builtin_amdgcn_wmma_f32_16x16x4_f32(
          false, a, false, bf, (short)0, acc[t], false, false);
    }
    aptr += 4;
    bptr += (size_t)4 * N;
  }

  // ---- epilogue: bias (+ReLU), store ----
#pragma unroll
  for (int t = 0; t < NT; ++t) {
    const int col = n + 16 * t;
    const float bv = bias[col];
#pragma unroll
    for (int r = 0; r < 8; ++r) {
      const int row = rowBase + r + 8 * halfSel;
      float v = acc[t][r] + bv;
      if (relu) v = fmaxf(v, 0.0f);
      C[(size_t)row * N + col] = v;
    }
  }
}

// ---------------------------------------------------------------------------
// Kernel 3: row softmax over L=256. One wave per row, 8 elems/lane.
// ---------------------------------------------------------------------------
__global__ __launch_bounds__(256) void softmax_rows_kernel(
    const float* __restrict__ logits, float* __restrict__ out) {
  const int wave = threadIdx.x >> 5, lane = threadIdx.x & 31;
  const int row = blockIdx.x * 8 + wave;
  const float* p = logits + (size_t)row * Lsz;

  float v[8];
  float mx = -3.4e38f;
#pragma unroll
  for (int i = 0; i < 8; ++i) { v[i] = p[lane + 32 * i]; mx = fmaxf(mx, v[i]); }
#pragma unroll
  for (int o = 16; o; o >>= 1) mx = fmaxf(mx, __shfl_xor(mx, o, 32));
  float s = 0.0f;
#pragma unroll
  for (int i = 0; i < 8; ++i) { v[i] = __expf(v[i] - mx); s += v[i]; }
#pragma unroll
  for (int o = 16; o; o >>= 1) s += __shfl_xor(s, o, 32);
  const float inv = 1.0f / s;
  float* q = out + (size_t)row * Lsz;
#pragma unroll
  for (int i = 0; i < 8; ++i) q[lane + 32 * i] = v[i] * inv;
}

// ---------------------------------------------------------------------------
// Kernel 4: attn_applied[b,:] = attn_weights[b,:] @ encoder_outputs[b,:,:].
// Streams the 1 GiB operand once (HBM-bound): block per b, 128 threads,
// float4/thread across H, weights in LDS, nontemporal loads (read-once data).
// ---------------------------------------------------------------------------
__global__ __launch_bounds__(128) void attn_apply_kernel(
    const float* __restrict__ wts /* B x L */,
    const float* __restrict__ enc /* B x L x H */,
    float* __restrict__ out /* B x H */) {
  __shared__ float w[Lsz];
  const int b = blockIdx.x, t = threadIdx.x;
  w[t]       = wts[(size_t)b * Lsz + t];
  w[t + 128] = wts[(size_t)b * Lsz + t + 128];
  __syncthreads();

  const v4f* e = (const v4f*)(enc + (size_t)b * Lsz * Hsz);
  v4f acc = {0.f, 0.f, 0.f, 0.f};
#pragma unroll 4
  for (int l = 0; l < Lsz; ++l) {
    const float wl = w[l];
    const v4f ev = __builtin_nontemporal_load(e + (size_t)l * 128 + t);
    acc[0] = fmaf(wl, ev[0], acc[0]);
    acc[1] = fmaf(wl, ev[1], acc[1]);
    acc[2] = fmaf(wl, ev[2], acc[2]);
    acc[3] = fmaf(wl, ev[3], acc[3]);
  }
  *(v4f*)(out + (size_t)b * Hsz + 4 * t) = acc;
}

// ---------------------------------------------------------------------------
// Kernel 5: GRU gates (elementwise over B*H).
// ---------------------------------------------------------------------------
__global__ __launch_bounds__(256) void gru_gate_kernel(
    const float* __restrict__ gi, const float* __restrict__ gh,
    const float* __restrict__ h0, float* __restrict__ hnew) {
  const int idx = blockIdx.x * 256 + threadIdx.x;  // 0 .. B*H-1
  const int b = idx >> 9, h = idx & (Hsz - 1);
  const float* gib = gi + (size_t)b * 3 * Hsz;
  const float* ghb = gh + (size_t)b * 3 * Hsz;
  const float ir = gib[h], iz = gib[Hsz + h], in_ = gib[2 * Hsz + h];
  const float hr = ghb[h], hz = ghb[Hsz + h], hn  = ghb[2 * Hsz + h];
  const float r = 1.0f / (1.0f + __expf(-(ir + hr)));
  const float z = 1.0f / (1.0f + __expf(-(iz + hz)));
  const float n = tanhf(in_ + r * hn);
  hnew[idx] = (1.0f - z) * n + z * h0[idx];
}

// ---------------------------------------------------------------------------
// Kernel 6: output[b] = dot(h_new[b,:], out_W[:,0]) + out_b. One wave per row.
// ---------------------------------------------------------------------------
__global__ __launch_bounds__(256) void out_proj_kernel(
    const float* __restrict__ hnew, const float* __restrict__ outW,
    const float* __restrict__ outb, float* __restrict__ out) {
  const int wave = threadIdx.x >> 5, lane = threadIdx.x & 31;
  const int b = blockIdx.x * 8 + wave;
  const float* hp = hnew + (size_t)b * Hsz;
  float s = 0.0f;
#pragma unroll
  for (int i = 0; i < 16; ++i) s = fmaf(hp[lane + 32 * i], outW[lane + 32 * i], s);
#pragma unroll
  for (int o = 16; o; o >>= 1) s += __shfl_xor(s, o, 32);
  if (lane == 0) out[b] = s + outb[0];
}

// ---------------------------------------------------------------------------
extern "C" void kernel_launch(void* const* d_in, const int* in_sizes, int n_in,
                              void* d_out, int out_size, void* d_ws, size_t ws_size,
                              hipStream_t stream) {
  (void)in_sizes; (void)n_in; (void)out_size; (void)ws_size;

  const float* x       = (const float*)d_in[0];   // (B,1,1)
  const float* hidden  = (const float*)d_in[1];   // (1,B,H)
  const float* enc     = (const float*)d_in[2];   // (B,L,H)
  const float* emb_W   = (const float*)d_in[3];   // (1,H)
  const float* emb_b   = (const float*)d_in[4];   // (H)
  const float* attn_W  = (const float*)d_in[5];   // (2H,L)
  const float* attn_b  = (const float*)d_in[6];   // (L)
  const float* comb_W  = (const float*)d_in[7];   // (2H,H)
  const float* comb_b  = (const float*)d_in[8];   // (H)
  const float* gru_Wih = (const float*)d_in[9];   // (H,3H)
  const float* gru_Whh = (const float*)d_in[10];  // (H,3H)
  const float* gru_bih = (const float*)d_in[11];  // (3H)
  const float* gru_bhh = (const float*)d_in[12];  // (3H)
  const float* out_W   = (const float*)d_in[13];  // (H,1)
  const float* out_b   = (const float*)d_in[14];  // (1)

  // Output tuple, flat: output (B) | h_new (B*H) | attn_weights (B*L)
  float* out0  = (float*)d_out;
  float* hnew  = out0 + Bsz;
  float* attnw = hnew + (size_t)Bsz * Hsz;

  // Workspace layout (fp32)
  float* ws       = (float*)d_ws;
  float* embedded = ws;                                   // B*H
  float* logits   = embedded + (size_t)Bsz * Hsz;         // B*L
  float* applied  = logits   + (size_t)Bsz * Lsz;         // B*H
  float* xt       = applied  + (size_t)Bsz * Hsz;         // B*H
  float* gi       = xt       + (size_t)Bsz * Hsz;         // B*3H
  float* gh       = gi       + (size_t)Bsz * 3 * Hsz;     // B*3H

  // 1) embed + layernorm
  embed_ln_kernel<<<Bsz, 128, 0, stream>>>(x, emb_W, emb_b, embedded);

  // 2) attn_logits = [embedded | hidden] @ attn_W + attn_b  (M=2048,K=1024,N=256)
  //    NT=2 -> 1024 waves (fills ~1024 SIMD32s)
  {
    dim3 grid(Lsz / 32, Bsz / 64);
    gemm_wmma_f32_kernel<2><<<grid, 128, 0, stream>>>(
        embedded, hidden, Hsz, Hsz, attn_W, attn_b, logits, Lsz, 0);
  }

  // 3) softmax -> attn_weights (output #3)
  softmax_rows_kernel<<<Bsz / 8, 256, 0, stream>>>(logits, attnw);

  // 4) attn_applied = attn_weights @ encoder_outputs  (streams 1 GiB)
  attn_apply_kernel<<<Bsz, 128, 0, stream>>>(attnw, enc, applied);

  // 5) xt = relu([embedded | attn_applied] @ comb_W + comb_b)  (N=512, NT=4)
  {
    dim3 grid(Hsz / 64, Bsz / 64);
    gemm_wmma_f32_kernel<4><<<grid, 128, 0, stream>>>(
        embedded, applied, Hsz, Hsz, comb_W, comb_b, xt, Hsz, 1);
  }

  // 6) gi = xt @ gru_Wih + gru_bih ; gh = h0 @ gru_Whh + gru_bhh  (N=1536, NT=4)
  {
    dim3 grid(3 * Hsz / 64, Bsz / 64);
    gemm_wmma_f32_kernel<4><<<grid, 128, 0, stream>>>(
        xt, xt, Hsz, 0, gru_Wih, gru_bih, gi, 3 * Hsz, 0);
    gemm_wmma_f32_kernel<4><<<grid, 128, 0, stream>>>(
        hidden, hidden, Hsz, 0, gru_Whh, gru_bhh, gh, 3 * Hsz, 0);
  }

  // 7) gates -> h_new (output #2)
  gru_gate_kernel<<<(Bsz * Hsz) / 256, 256, 0, stream>>>(gi, gh, hidden, hnew);

  // 8) output = h_new @ out_W + out_b (output #1)
  out_proj_kernel<<<Bsz / 8, 256, 0, stream>>>(hnew, out_W, out_b, out0);
}